// TripletLoss_55181739819242
// MI455X (gfx1250) — compile-verified
//
#include <hip/hip_runtime.h>
#include <hip/hip_bf16.h>

#define NN 8192
#define DD 512
#define NCLS 64
#define NTILES (NN / 16)     // 512 tiles per side
#define LDS_STRIDE 516       // 512 + 4 pad: lane stride of 516 dwords -> 16 distinct LDS banks

typedef float v2f __attribute__((ext_vector_type(2)));
typedef float v8f __attribute__((ext_vector_type(8)));

// ---------------- kernel 0: zero scratch region ----------------
__global__ void tl_zero_kernel(float* p, int n) {
    int i = blockIdx.x * blockDim.x + threadIdx.x;
    if (i < n) p[i] = 0.0f;
}

// ---------------- kernel 1: row squared norms (wave per row) ----------------
__global__ __launch_bounds__(256) void tl_rownorm_kernel(const float* __restrict__ z,
                                                         float* __restrict__ sq) {
    const int wave = (blockIdx.x * 256 + threadIdx.x) >> 5;   // wave32 -> one row per wave
    const int lane = threadIdx.x & 31;
    if (wave >= NN) return;
    const float* row = z + (size_t)wave * DD;
    float s = 0.0f;
    for (int c = lane * 4; c < DD; c += 32 * 4) {
        float4 v = *(const float4*)(row + c);
        s += v.x * v.x + v.y * v.y + v.z * v.z + v.w * v.w;
    }
    for (int m = 16; m >= 1; m >>= 1) s += __shfl_xor(s, m, 32);
    if (lane == 0) sq[wave] = s;
}

// ---------------- kernel 2: class histogram (single block) ----------------
__global__ void tl_hist_kernel(const int* __restrict__ labels, int* __restrict__ classCnt) {
    __shared__ int h[NCLS];
    if (threadIdx.x < NCLS) h[threadIdx.x] = 0;
    __syncthreads();
    for (int i = threadIdx.x; i < NN; i += blockDim.x) atomicAdd(&h[labels[i]], 1);
    __syncthreads();
    if (threadIdx.x < NCLS) classCnt[threadIdx.x] = h[threadIdx.x];
}

// ---------------- WMMA tile processor: NT tiles (stride 8 tiles apart) ----------------
// All B addresses are derived from ONE base pointer + compile-time immediate offsets
// (tile stride 8*16*DD floats = 262144 B < 24-bit VMEM immediate range) so the
// compiler can clause the loads and use staggered s_wait_loadcnt pipelining.
template <int NT>
__device__ __forceinline__ void tl_process(const float* __restrict__ z,
                                           const int* __restrict__ labels,
                                           const float* __restrict__ sq,
                                           float* __restrict__ pos_sum,
                                           float* __restrict__ neg_sum,
                                           const float* arow_lane,   // arow + l15*LDS_STRIDE + 2*hi
                                           int ctile0, int rowTile,
                                           int l15, int hi,
                                           const float* sqr, const int* labr,
                                           float* posAcc, float* negAcc) {
    constexpr int TSTR = 8 * 16 * DD;   // floats between consecutive tiles of this wave

    const float* pbase = z + (size_t)(ctile0 * 16 + l15) * DD + 2 * hi;

    v8f acc[NT];
#pragma unroll
    for (int t = 0; t < NT; ++t) acc[t] = v8f{};

#pragma unroll 4
    for (int k = 0; k < DD; k += 4) {
        // A fragment: 16x4 fp32, lane&15 = M, VGPR pair = K {0,1}/{2,3} by half-wave
        v2f a = *(const v2f*)(arow_lane + k);
        // B fragments: 4x16, N = lane&15, same K striping; constant immediate offsets
        v2f b[NT];
#pragma unroll
        for (int t = 0; t < NT; ++t) b[t] = *(const v2f*)(pbase + k + t * TSTR);
#pragma unroll
        for (int t = 0; t < NT; ++t)
            acc[t] = __builtin_amdgcn_wmma_f32_16x16x4_f32(false, a, false, b[t],
                                                           (short)0, acc[t], false, false);
    }

#pragma unroll
    for (int t = 0; t < NT; ++t) {
        const int   ctile = ctile0 + 8 * t;
        const int   col   = ctile * 16 + l15;
        const float sc    = sq[col];
        const int   lc    = labels[col];
        float cp = 0.0f, cn = 0.0f;
#pragma unroll
        for (int v = 0; v < 8; ++v) {
            float d2   = sqr[v] + sc - 2.0f * acc[t][v];
            float dist = (d2 > 0.0f) ? __builtin_sqrtf(d2) : 0.0f;
            bool  same = (labr[v] == lc);
            float pd = same ? dist : 0.0f;
            float nd = same ? 0.0f : dist;
            posAcc[v] += pd;    // row-side (row = rowBase + v + 8*hi)
            negAcc[v] += nd;
            cp += pd;           // column-side partial over this lane's 8 rows
            cn += nd;
        }
        // Fold the two half-waves sharing the same column N = l15.
        cp += __shfl_xor(cp, 16, 32);
        cn += __shfl_xor(cn, 16, 32);
        if (ctile != rowTile) {              // diagonal tile already covers both orders
            if (hi == 0) {
                atomicAdd(&pos_sum[col], cp);
                atomicAdd(&neg_sum[col], cn);
            }
        }
    }
}

// ---------------- kernel 3: symmetric fused WMMA gram -> distance -> masked sums ----
// Block = one 16-row tile (staged in LDS). Wave w covers column tiles c >= rowTile
// with c = rowTile + w + 8*i  (upper triangle only; each off-diagonal tile feeds
// both its rows and, via transpose, its columns).
__global__ __launch_bounds__(256) void tl_gram_dist_kernel(const float* __restrict__ z,
                                                           const int* __restrict__ labels,
                                                           const float* __restrict__ sq,
                                                           float* __restrict__ pos_sum,
                                                           float* __restrict__ neg_sum) {
    __shared__ float arow[16 * LDS_STRIDE];

    const int rowTile = blockIdx.x;            // 0..511
    const int wave    = threadIdx.x >> 5;      // 0..7
    const int lane    = threadIdx.x & 31;
    const int hi      = lane >> 4;             // half-wave 0/1
    const int l15     = lane & 15;
    const int rowBase = rowTile * 16;

    // Stage the 16 x 512 A row block into LDS (padded stride).
    for (int idx = threadIdx.x; idx < 16 * (DD / 4); idx += 256) {
        int r  = idx / (DD / 4);
        int c4 = (idx % (DD / 4)) * 4;
        float4 v = *(const float4*)(z + (size_t)(rowBase + r) * DD + c4);
        *(float4*)(arow + r * LDS_STRIDE + c4) = v;
    }
    __syncthreads();

    // Per-lane row metadata: this lane owns rows rowBase + v + 8*hi, v = 0..7.
    float sqr[8];
    int   labr[8];
#pragma unroll
    for (int v = 0; v < 8; ++v) {
        int r = rowBase + v + 8 * hi;
        sqr[v]  = sq[r];
        labr[v] = labels[r];
    }

    float posAcc[8], negAcc[8];
#pragma unroll
    for (int v = 0; v < 8; ++v) { posAcc[v] = 0.0f; negAcc[v] = 0.0f; }

    const float* arow_lane = arow + l15 * LDS_STRIDE + 2 * hi;

    // Column-tile list for this wave: first, first+8, first+16, ...
    const int first  = rowTile + wave;
    const int ntiles = (first <= NTILES - 1) ? (((NTILES - 1) - first) >> 3) + 1 : 0;

    int i = 0;
    for (; i + 4 <= ntiles; i += 4)            // groups of 4: A reuse x4, 4 indep accumulators
        tl_process<4>(z, labels, sq, pos_sum, neg_sum, arow_lane,
                      first + 8 * i, rowTile, l15, hi, sqr, labr, posAcc, negAcc);
    for (; i < ntiles; ++i)                    // tail (<= 3 tiles)
        tl_process<1>(z, labels, sq, pos_sum, neg_sum, arow_lane,
                      first + 8 * i, rowTile, l15, hi, sqr, labr, posAcc, negAcc);

    // Row-side flush: reduce across the 16 lanes of each half-wave, 2 atomics per row.
#pragma unroll
    for (int v = 0; v < 8; ++v) {
        float p = posAcc[v], n = negAcc[v];
        for (int m = 8; m >= 1; m >>= 1) {
            p += __shfl_xor(p, m, 16);
            n += __shfl_xor(n, m, 16);
        }
        if (l15 == 0) {
            int r = rowBase + v + 8 * hi;
            atomicAdd(&pos_sum[r], p);
            atomicAdd(&neg_sum[r], n);
        }
    }
}

// ---------------- kernel 4: per-row loss, reduce to totals ----------------
__global__ void tl_rowfinal_kernel(const float* __restrict__ pos_sum,
                                   const float* __restrict__ neg_sum,
                                   const int* __restrict__ labels,
                                   const int* __restrict__ classCnt,
                                   float* __restrict__ totals) {
    int i = blockIdx.x * blockDim.x + threadIdx.x;
    if (i >= NN) return;
    int pc = classCnt[labels[i]];
    int nc = NN - pc;
    float pm = pos_sum[i] / (float)(pc > 1 ? pc : 1);
    float nm = neg_sum[i] / (float)(nc > 1 ? nc : 1);
    bool valid = (pc > 1) && (nc > 0);
    float pr = pm - nm + 1.0f;           // MARGIN = 1.0
    pr = pr > 0.0f ? pr : 0.0f;
    if (valid) {
        atomicAdd(&totals[0], pr);
        atomicAdd(&totals[1], 1.0f);
    }
}

// ---------------- kernel 5: final scalar ----------------
__global__ void tl_finalize_kernel(const float* __restrict__ totals, float* __restrict__ out) {
    out[0] = totals[0] / fmaxf(totals[1], 1.0f);
}

extern "C" void kernel_launch(void* const* d_in, const int* in_sizes, int n_in,
                              void* d_out, int out_size, void* d_ws, size_t ws_size,
                              hipStream_t stream) {
    const float* z      = (const float*)d_in[0];
    const int*   labels = (const int*)d_in[1];
    float* ws  = (float*)d_ws;
    float* out = (float*)d_out;

    // Workspace layout (float units):
    // [0, 8192)      sq
    // [8192, 16384)  pos_sum
    // [16384, 24576) neg_sum
    // [24576, 24578) totals {total, count}
    // [24580, 24644) classCnt (int)
    float* sq       = ws;
    float* pos_sum  = ws + 8192;
    float* neg_sum  = ws + 16384;
    float* totals   = ws + 24576;
    int*   classCnt = (int*)(ws + 24580);

    // Zero pos/neg/totals/classCnt region: [8192, 24644) -> 16452 floats.
    tl_zero_kernel<<<(16452 + 255) / 256, 256, 0, stream>>>(pos_sum, 16452);

    // Row squared norms: 8192 rows, 8 waves/block.
    tl_rownorm_kernel<<<NN / 8, 256, 0, stream>>>(z, sq);

    // Class histogram.
    tl_hist_kernel<<<1, 256, 0, stream>>>(labels, classCnt);

    // Symmetric fused WMMA gram + distance + masked sums: 512 row-tile blocks,
    // each covering only column tiles c >= rowTile (upper triangle, half the FLOPs).
    tl_gram_dist_kernel<<<NTILES, 256, 0, stream>>>(z, labels, sq, pos_sum, neg_sum);

    // Per-row loss + global reduction.
    tl_rowfinal_kernel<<<NN / 256, 256, 0, stream>>>(pos_sum, neg_sum, labels, classCnt, totals);

    // Final scalar.
    tl_finalize_kernel<<<1, 1, 0, stream>>>(totals, out);
}